// Sequence_59554016526398
// MI455X (gfx1250) — compile-verified
//
#include <hip/hip_runtime.h>
#include <hip/hip_bf16.h>

#define VOCAB 20
#define EMBED 30
#define H1 40
#define H2 50
#define G1 (4*H1)   // 160
#define G2 (4*H2)   // 200
#define SEQ 500000

typedef float v2f __attribute__((ext_vector_type(2)));
typedef float v8f __attribute__((ext_vector_type(8)));

__device__ __forceinline__ float sigmoid_f(float x) {
    return 1.0f / (1.0f + __expf(-x));
}
__device__ __forceinline__ float tanh_f(float x) {
    // tanh(x) = 1 - 2/(e^{2x}+1); __expf saturates cleanly at +/-inf
    return 1.0f - 2.0f / (__expf(2.0f * x) + 1.0f);
}

// ---------------------------------------------------------------------------
// Kernel 1: single-workgroup sequential 2-layer LSTM.
//   - Per-thread gate weights live in REGISTERS for all 500k steps.
//   - LDS holds only the token->gate table P (x@Wih1^T + biases), the h
//     state vectors (read as float4 broadcasts), and activated gates.
//   - Wave-aligned thread groups: waves 0-4 compute layer-1 gates while
//     waves 5-11 compute the layer-2 h2.Whh2^T partial in parallel.
//   - c1/c2 cell state in registers; h2_t streamed to d_ws for the WMMA
//     projection kernel.
// ---------------------------------------------------------------------------
extern "C" __global__ __launch_bounds__(384, 1)
void lstm_seq_kernel(const int*   __restrict__ tokens,
                     const float* __restrict__ embed_w,
                     const float* __restrict__ w_ih1, const float* __restrict__ w_hh1,
                     const float* __restrict__ b_ih1, const float* __restrict__ b_hh1,
                     const float* __restrict__ w_ih2, const float* __restrict__ w_hh2,
                     const float* __restrict__ b_ih2, const float* __restrict__ b_hh2,
                     float* __restrict__ h2_all)
{
    __shared__ __align__(16) float P[VOCAB * G1];   // 12.8 KB
    __shared__ __align__(16) float h1s[H1];
    __shared__ __align__(16) float h2s[H2];
    __shared__ __align__(16) float gates1[G1];
    __shared__ __align__(16) float gates2[G2];

    const int tid = threadIdx.x;

    // ---- one-time setup (amortized over 500k steps) ----
    for (int idx = tid; idx < VOCAB * G1; idx += 384) {
        const int v = idx / G1, g = idx % G1;
        float acc = b_ih1[g] + b_hh1[g];
        #pragma unroll
        for (int e = 0; e < EMBED; ++e)
            acc = __builtin_fmaf(embed_w[v * EMBED + e], w_ih1[g * EMBED + e], acc);
        P[idx] = acc;
    }
    if (tid < H1) h1s[tid] = 0.0f;
    if (tid < H2) h2s[tid] = 0.0f;

    // ---- hoist this thread's weight rows into registers ----
    float w1r[H1];            // layer-1 hh row   (waves 0-4:  tid < 160)
    float w2h[H2];            // layer-2 hh row   (waves 5-11: 160 <= tid < 360)
    float w2i[H1];            // layer-2 ih row
    float b2r = 0.0f;
    if (tid < G1) {
        #pragma unroll
        for (int k = 0; k < H1; ++k) w1r[k] = w_hh1[tid * H1 + k];
    } else if (tid < G1 + G2) {
        const int g2 = tid - G1;
        #pragma unroll
        for (int k = 0; k < H2; ++k) w2h[k] = w_hh2[g2 * H2 + k];
        #pragma unroll
        for (int k = 0; k < H1; ++k) w2i[k] = w_ih2[g2 * H1 + k];
        b2r = b_ih2[g2] + b_hh2[g2];
    }
    float c1 = 0.0f, c2 = 0.0f;
    __syncthreads();

    float acc2 = 0.0f;        // layer-2 hh partial, carried across barriers
    for (int t = 0; t < SEQ; ++t) {
        const int tok = tokens[t];
        {
            const int tp = (t + 256 < SEQ) ? (t + 256) : (SEQ - 1);
            __builtin_prefetch(&tokens[tp], 0, 0);
        }

        // Phase 1 (wave-parallel groups)
        if (tid < G1) {
            float acc = P[tok * G1 + tid];
            #pragma unroll
            for (int k = 0; k < H1; k += 4) {
                const float4 h4 = *(const float4*)&h1s[k];   // LDS b128 broadcast
                acc = __builtin_fmaf(h4.x, w1r[k],     acc);
                acc = __builtin_fmaf(h4.y, w1r[k + 1], acc);
                acc = __builtin_fmaf(h4.z, w1r[k + 2], acc);
                acc = __builtin_fmaf(h4.w, w1r[k + 3], acc);
            }
            gates1[tid] = (tid >= 2 * H1 && tid < 3 * H1) ? tanh_f(acc) : sigmoid_f(acc);
        } else if (tid < G1 + G2) {
            float a = b2r;
            #pragma unroll
            for (int k = 0; k < 48; k += 4) {
                const float4 h4 = *(const float4*)&h2s[k];
                a = __builtin_fmaf(h4.x, w2h[k],     a);
                a = __builtin_fmaf(h4.y, w2h[k + 1], a);
                a = __builtin_fmaf(h4.z, w2h[k + 2], a);
                a = __builtin_fmaf(h4.w, w2h[k + 3], a);
            }
            a = __builtin_fmaf(h2s[48], w2h[48], a);
            a = __builtin_fmaf(h2s[49], w2h[49], a);
            acc2 = a;
        }
        __syncthreads();

        // Phase 2: layer-1 cell/hidden update
        if (tid < H1) {
            const float ig = gates1[tid];
            const float fg = gates1[H1 + tid];
            const float gg = gates1[2 * H1 + tid];
            const float og = gates1[3 * H1 + tid];
            c1 = __builtin_fmaf(fg, c1, ig * gg);
            h1s[tid] = og * tanh_f(c1);
        }
        __syncthreads();

        // Phase 3: layer-2 ih dots + activation
        if (tid >= G1 && tid < G1 + G2) {
            const int g2 = tid - G1;
            float a = acc2;
            #pragma unroll
            for (int k = 0; k < H1; k += 4) {
                const float4 h4 = *(const float4*)&h1s[k];
                a = __builtin_fmaf(h4.x, w2i[k],     a);
                a = __builtin_fmaf(h4.y, w2i[k + 1], a);
                a = __builtin_fmaf(h4.z, w2i[k + 2], a);
                a = __builtin_fmaf(h4.w, w2i[k + 3], a);
            }
            gates2[g2] = (g2 >= 2 * H2 && g2 < 3 * H2) ? tanh_f(a) : sigmoid_f(a);
        }
        __syncthreads();

        // Phase 4: layer-2 cell/hidden update + stream h2_t to global
        if (tid < H2) {
            const float ig = gates2[tid];
            const float fg = gates2[H2 + tid];
            const float gg = gates2[2 * H2 + tid];
            const float og = gates2[3 * H2 + tid];
            c2 = __builtin_fmaf(fg, c2, ig * gg);
            const float h2v = og * tanh_f(c2);
            h2s[tid] = h2v;
            h2_all[(long)t * H2 + tid] = h2v;
        }
        __syncthreads();
    }
}

// ---------------------------------------------------------------------------
// Kernel 2: out = h2_all[T,50] @ lin_w^T[50,20] + lin_b via
// v_wmma_f32_16x16x4_f32. 320 threads = 10 waves; each wave owns one 16-row
// tile (160 rows/block, 3125 blocks). A tile + zero-padded B staged in LDS:
//   - A staged with fully coalesced contiguous global reads, pad K 50->52
//   - B packed as float2 K-pairs (26 x 32), pad N 20->32, K 50->52
// so every fragment read is an unconditional conflict-free ds_load_b64.
// ---------------------------------------------------------------------------
#define OP_WAVES   10
#define OP_THREADS (OP_WAVES * 32)
#define OP_ROWS    (OP_WAVES * 16)      // 160
#define OP_K       52                   // padded K
#define OP_N       32                   // padded N

extern "C" __global__ __launch_bounds__(OP_THREADS, 2)
void outproj_wmma_kernel(const float* __restrict__ h2_all,
                         const float* __restrict__ lin_w,
                         const float* __restrict__ lin_b,
                         float* __restrict__ out)
{
    __shared__ __align__(16) float Asub[OP_ROWS * OP_K];   // 33.3 KB
    __shared__ __align__(16) v2f   Bp[(OP_K / 2) * OP_N];  // 6.7 KB, K-pair packed
    __shared__ __align__(16) float bias[OP_N];

    const int  tid     = threadIdx.x;
    const long rowBase = (long)blockIdx.x * OP_ROWS;

    // stage padded B: Bp[j][n] = { B[2j][n], B[2j+1][n] },  B[k][n] = lin_w[n][k]
    for (int i = tid; i < (OP_K / 2) * OP_N; i += OP_THREADS) {
        const int j = i / OP_N, n = i % OP_N;
        const int k0 = 2 * j;
        v2f b;
        b.x = (n < VOCAB && k0     < H2) ? lin_w[n * H2 + k0]     : 0.0f;
        b.y = (n < VOCAB && k0 + 1 < H2) ? lin_w[n * H2 + k0 + 1] : 0.0f;
        Bp[i] = b;
    }
    for (int i = tid; i < OP_N; i += OP_THREADS)
        bias[i] = (i < VOCAB) ? lin_b[i] : 0.0f;

    // stage A: coalesced contiguous read of 160x50 block, LDS stride 52
    const float* Ag = h2_all + rowBase * H2;
    for (int i = tid; i < OP_ROWS * H2; i += OP_THREADS) {
        const int r = i / H2, c = i % H2;
        Asub[r * OP_K + c] = Ag[i];
    }
    for (int i = tid; i < OP_ROWS * 2; i += OP_THREADS)    // zero pad k = 50,51
        Asub[(i >> 1) * OP_K + H2 + (i & 1)] = 0.0f;
    __syncthreads();

    const int wave = tid >> 5;
    const int lane = tid & 31;
    const int half = lane >> 4;
    const int m16  = lane & 15;
    const float* Arow = &Asub[(wave * 16 + m16) * OP_K];

    v8f c0 = {};
    v8f c1 = {};
    #pragma unroll
    for (int kk = 0; kk < 13; ++kk) {
        const int k0 = kk * 4 + 2 * half;       // fragment K base for this half
        const int j  = kk * 2 + half;           // K-pair index
        const v2f a  = *(const v2f*)&Arow[k0];  // ds_load_b64
        const v2f b0 = Bp[j * OP_N + m16];      // ds_load_b64
        const v2f b1 = Bp[j * OP_N + 16 + m16]; // ds_load_b64
        c0 = __builtin_amdgcn_wmma_f32_16x16x4_f32(false, a, false, b0,
                                                   (short)0, c0, false, false);
        c1 = __builtin_amdgcn_wmma_f32_16x16x4_f32(false, a, false, b1,
                                                   (short)0, c1, false, false);
    }

    #pragma unroll
    for (int r = 0; r < 8; ++r) {
        const long m  = rowBase + wave * 16 + r + 8 * half;
        out[m * VOCAB + m16] = c0[r] + bias[m16];
        const int n1 = 16 + m16;
        if (n1 < VOCAB)
            out[m * VOCAB + n1] = c1[r] + bias[n1];
    }
}

// ---------------------------------------------------------------------------
extern "C" void kernel_launch(void* const* d_in, const int* in_sizes, int n_in,
                              void* d_out, int out_size, void* d_ws, size_t ws_size,
                              hipStream_t stream)
{
    (void)in_sizes; (void)n_in; (void)out_size; (void)ws_size;

    const int*   tokens  = (const int*)  d_in[0];
    const float* embed_w = (const float*)d_in[1];
    const float* w_ih1   = (const float*)d_in[2];
    const float* w_hh1   = (const float*)d_in[3];
    const float* b_ih1   = (const float*)d_in[4];
    const float* b_hh1   = (const float*)d_in[5];
    const float* w_ih2   = (const float*)d_in[6];
    const float* w_hh2   = (const float*)d_in[7];
    const float* b_ih2   = (const float*)d_in[8];
    const float* b_hh2   = (const float*)d_in[9];
    const float* lin_w   = (const float*)d_in[10];
    const float* lin_b   = (const float*)d_in[11];
    float* out    = (float*)d_out;
    float* h2_all = (float*)d_ws;   // SEQ*H2 f32 = 100 MB scratch

    hipLaunchKernelGGL(lstm_seq_kernel, dim3(1), dim3(384), 0, stream,
                       tokens, embed_w, w_ih1, w_hh1, b_ih1, b_hh1,
                       w_ih2, w_hh2, b_ih2, b_hh2, h2_all);

    hipLaunchKernelGGL(outproj_wmma_kernel, dim3(SEQ / OP_ROWS), dim3(OP_THREADS),
                       0, stream, h2_all, lin_w, lin_b, out);
}